// GCN_2542620639751
// MI455X (gfx1250) — compile-verified
//
#include <hip/hip_runtime.h>
#include <hip/hip_bf16.h>

typedef __attribute__((ext_vector_type(2))) float v2f;
typedef __attribute__((ext_vector_type(8))) float v8f;

// ---------- helpers ----------
__device__ __forceinline__ unsigned floatFlip(float f) {
  unsigned u = __float_as_uint(f);
  unsigned mask = (unsigned)(-(int)(u >> 31)) | 0x80000000u;
  return u ^ mask;                       // monotone: float order -> unsigned order
}
__device__ __forceinline__ float floatUnflip(unsigned u) {
  unsigned mask = ((u >> 31) - 1u) | 0x80000000u;
  return __uint_as_float(u ^ mask);
}
__device__ __forceinline__ float lrelu(float v) { return v > 0.f ? v : 0.2f * v; }

__device__ __forceinline__ void edge_sd(const int* __restrict__ ei, int e, int E,
                                        int& s, int& d) {
  if (e < E) { s = ei[e]; d = ei[E + e]; }
  else       { s = e - E; d = e - E; }        // appended self loops
}

// ---------- per-layer init ----------
__global__ void init_nodes(unsigned* __restrict__ mEnc, float* __restrict__ den,
                           float* __restrict__ acc, int Fo, int n) {
  int i = blockIdx.x * blockDim.x + threadIdx.x;
  if (i >= n) return;
  mEnc[i] = floatFlip(-__builtin_inff());
  den[i] = 0.f;
  for (int f = 0; f < Fo; ++f) acc[i * Fo + f] = 0.f;
}

// ---------- H = X (n x K) @ W (K x 16) via V_WMMA_F32_16X16X4_F32 ----------
// One wave32 computes one 16x16 output tile; chain K/4 wmma ops (f32 A/B/C).
// A 16x4 f32 layout: lanes 0-15 row M=lane hold K={0,1}; lanes 16-31 hold K={2,3}.
// B 4x16 layout mirrored: lane half selects K pair, lane&15 selects column N.
template <int K>
__global__ __launch_bounds__(32) void gemm16_wmma(const float* __restrict__ X,
                                                  const float* __restrict__ W,
                                                  float* __restrict__ H, int n) {
  const int tile = blockIdx.x;
  const int lane = threadIdx.x & 31;
  const int ln   = lane & 15;
  const int kh   = (lane >> 4) << 1;          // 0 (lanes 0-15) or 2 (lanes 16-31)
  int row = tile * 16 + ln;
  if (row >= n) row = n - 1;                  // clamp reads; EXEC stays all-ones
  v8f c = {};
#pragma unroll
  for (int kk = 0; kk < K; kk += 4) {
    v2f a, b;
    a.x = X[(size_t)row * K + kk + kh];
    a.y = X[(size_t)row * K + kk + kh + 1];
    b.x = W[(size_t)(kk + kh) * 16 + ln];
    b.y = W[(size_t)(kk + kh + 1) * 16 + ln];
    c = __builtin_amdgcn_wmma_f32_16x16x4_f32(false, a, false, b, (short)0, c,
                                              false, false);
  }
  const int rbase = tile * 16 + (lane >> 4) * 8;
  if (tile * 16 + 16 <= n) {
    // fast path: whole tile in range -> straight-line stores, no exec juggling
#pragma unroll
    for (int v = 0; v < 8; ++v)
      H[(size_t)(rbase + v) * 16 + ln] = c[v];
  } else {
#pragma unroll
    for (int v = 0; v < 8; ++v) {
      int r = rbase + v;
      if (r < n) H[(size_t)r * 16 + ln] = c[v];
    }
  }
}

// ---------- per-node attention logits ----------
__global__ void node_scores16(const float* __restrict__ H, const float* __restrict__ a_s,
                              const float* __restrict__ a_d, float* __restrict__ es,
                              float* __restrict__ ed, int n) {
  int i = blockIdx.x * blockDim.x + threadIdx.x;
  if (i >= n) return;
  float s = 0.f, d = 0.f;
#pragma unroll
  for (int f = 0; f < 16; ++f) {
    float hv = H[(size_t)i * 16 + f];
    s += hv * a_s[f];
    d += hv * a_d[f];
  }
  es[i] = s; ed[i] = d;
}

// p-head: hp = X32 @ Wp (32x1), plus its scalar logits
__global__ void node_hp(const float* __restrict__ X32, const float* __restrict__ Wp,
                        const float* __restrict__ asp, const float* __restrict__ adp,
                        float* __restrict__ hp, float* __restrict__ es,
                        float* __restrict__ ed, int n) {
  int i = blockIdx.x * blockDim.x + threadIdx.x;
  if (i >= n) return;
  float s = 0.f;
#pragma unroll
  for (int k = 0; k < 32; ++k) s += X32[(size_t)i * 32 + k] * Wp[k];
  hp[i] = s;
  es[i] = s * asp[0];
  ed[i] = s * adp[0];
}

// ---------- edge-parallel segment softmax ----------
__global__ void edge_max(const int* __restrict__ ei, int E, int n,
                         const float* __restrict__ es, const float* __restrict__ ed,
                         unsigned* __restrict__ mEnc) {
  int e = blockIdx.x * blockDim.x + threadIdx.x;
  if (e >= E + n) return;
  int s, d; edge_sd(ei, e, E, s, d);
  atomicMax(&mEnc[d], floatFlip(lrelu(es[s] + ed[d])));
}

__global__ void edge_sum(const int* __restrict__ ei, int E, int n,
                         const float* __restrict__ es, const float* __restrict__ ed,
                         const unsigned* __restrict__ mEnc, float* __restrict__ den) {
  int e = blockIdx.x * blockDim.x + threadIdx.x;
  if (e >= E + n) return;
  int s, d; edge_sd(ei, e, E, s, d);
  float v = lrelu(es[s] + ed[d]);
  atomicAdd(&den[d], __expf(v - floatUnflip(mEnc[d])));
}

__global__ void edge_scatter16(const int* __restrict__ ei, int E, int n,
                               const float* __restrict__ es, const float* __restrict__ ed,
                               const unsigned* __restrict__ mEnc,
                               const float* __restrict__ den,
                               const float* __restrict__ H, float* __restrict__ acc) {
  int e = blockIdx.x * blockDim.x + threadIdx.x;
  if (e >= E + n) return;
  int s, d; edge_sd(ei, e, E, s, d);
  float v = lrelu(es[s] + ed[d]);
  float alpha = __expf(v - floatUnflip(mEnc[d])) / (den[d] + 1e-16f);
  const float4* hs = (const float4*)(H + (size_t)s * 16);
  float4 h0 = hs[0], h1 = hs[1], h2 = hs[2], h3 = hs[3];
  float* a = acc + (size_t)d * 16;
  atomicAdd(a + 0,  h0.x * alpha); atomicAdd(a + 1,  h0.y * alpha);
  atomicAdd(a + 2,  h0.z * alpha); atomicAdd(a + 3,  h0.w * alpha);
  atomicAdd(a + 4,  h1.x * alpha); atomicAdd(a + 5,  h1.y * alpha);
  atomicAdd(a + 6,  h1.z * alpha); atomicAdd(a + 7,  h1.w * alpha);
  atomicAdd(a + 8,  h2.x * alpha); atomicAdd(a + 9,  h2.y * alpha);
  atomicAdd(a + 10, h2.z * alpha); atomicAdd(a + 11, h2.w * alpha);
  atomicAdd(a + 12, h3.x * alpha); atomicAdd(a + 13, h3.y * alpha);
  atomicAdd(a + 14, h3.z * alpha); atomicAdd(a + 15, h3.w * alpha);
}

__global__ void edge_scatter1(const int* __restrict__ ei, int E, int n,
                              const float* __restrict__ es, const float* __restrict__ ed,
                              const unsigned* __restrict__ mEnc,
                              const float* __restrict__ den,
                              const float* __restrict__ hp, float* __restrict__ acc) {
  int e = blockIdx.x * blockDim.x + threadIdx.x;
  if (e >= E + n) return;
  int s, d; edge_sd(ei, e, E, s, d);
  float v = lrelu(es[s] + ed[d]);
  float alpha = __expf(v - floatUnflip(mEnc[d])) / (den[d] + 1e-16f);
  atomicAdd(&acc[d], hp[s] * alpha);
}

// ---------- relu+bias, build [xa | x] concat for next layer ----------
__global__ void finalize16(const float* __restrict__ acc, const float* __restrict__ b,
                           const float* __restrict__ x, float* __restrict__ in2, int n) {
  int i = blockIdx.x * blockDim.x + threadIdx.x;
  if (i >= n) return;
#pragma unroll
  for (int f = 0; f < 16; ++f) {
    float v = acc[(size_t)i * 16 + f] + b[f];
    in2[(size_t)i * 32 + f] = v > 0.f ? v : 0.f;
    in2[(size_t)i * 32 + 16 + f] = x[(size_t)i * 16 + f];
  }
}

__global__ void finalize_p(const float* __restrict__ acc, const float* __restrict__ bp,
                           float* __restrict__ out, int n) {
  int i = blockIdx.x * blockDim.x + threadIdx.x;
  if (i >= n) return;
  float v = acc[i] + bp[0];
  out[i] = v > 0.f ? v : 0.f;
}

// ---------- value head: mean-pool xa3 (in2 cols 0..15) per graph + 2-layer MLP ----------
__global__ __launch_bounds__(256) void value_head(const float* __restrict__ in2,
                                                  const float* __restrict__ fc1w,
                                                  const float* __restrict__ fc1b,
                                                  const float* __restrict__ fc2w,
                                                  const float* __restrict__ fc2b,
                                                  float* __restrict__ out, int P) {
  __shared__ float sh[16];
  const int g = blockIdx.x;
  const int t = threadIdx.x;
  if (t < 16) sh[t] = 0.f;
  __syncthreads();
  float loc[16];
#pragma unroll
  for (int f = 0; f < 16; ++f) loc[f] = 0.f;
  for (int i = t; i < P; i += 256) {
    const float* row = in2 + (size_t)(g * P + i) * 32;
#pragma unroll
    for (int f = 0; f < 16; ++f) loc[f] += row[f];
  }
#pragma unroll
  for (int f = 0; f < 16; ++f) atomicAdd(&sh[f], loc[f]);   // ds_add_f32
  __syncthreads();
  if (t == 0) {
    float v[16];
#pragma unroll
    for (int f = 0; f < 16; ++f) v[f] = sh[f] / (float)P;
    float o = fc2b[0];
#pragma unroll
    for (int j = 0; j < 16; ++j) {
      float hj = fc1b[j];
#pragma unroll
      for (int f = 0; f < 16; ++f) hj += v[f] * fc1w[f * 16 + j];
      hj = hj > 0.f ? hj : 0.f;
      o += hj * fc2w[j];
    }
    out[g] = o;
  }
}

// ---------- host side ----------
extern "C" void kernel_launch(void* const* d_in, const int* in_sizes, int n_in,
                              void* d_out, int out_size, void* d_ws, size_t ws_size,
                              hipStream_t stream) {
  const float* x   = (const float*)d_in[0];
  const int*   ei  = (const int*)d_in[1];
  // d_in[2] edge_attr: ignored by GATConv (edge_dim=None)
  const float* W1  = (const float*)d_in[3];
  const float* as1 = (const float*)d_in[4];
  const float* ad1 = (const float*)d_in[5];
  const float* b1  = (const float*)d_in[6];
  const float* W2  = (const float*)d_in[7];
  const float* as2 = (const float*)d_in[8];
  const float* ad2 = (const float*)d_in[9];
  const float* b2  = (const float*)d_in[10];
  const float* W3  = (const float*)d_in[11];
  const float* as3 = (const float*)d_in[12];
  const float* ad3 = (const float*)d_in[13];
  const float* b3  = (const float*)d_in[14];
  const float* Wp  = (const float*)d_in[15];
  const float* asp = (const float*)d_in[16];
  const float* adp = (const float*)d_in[17];
  const float* bp  = (const float*)d_in[18];
  const float* f1w = (const float*)d_in[19];
  const float* f1b = (const float*)d_in[20];
  const float* f2w = (const float*)d_in[21];
  const float* f2b = (const float*)d_in[22];

  const int N = in_sizes[0] / 16;     // 100000
  const int E = in_sizes[1] / 2;      // 3200000
  const int G = out_size - N;         // graphs (100): out = N px + G vx
  const int P = (G > 0) ? N / G : N;  // nodes per graph (1000)

  float* ws      = (float*)d_ws;
  float*    H    = ws;                       // N*16  (also reused as hp for p-head)
  float*    es   = H + (size_t)N * 16;       // N
  float*    ed   = es + N;                   // N
  unsigned* mEnc = (unsigned*)(ed + N);      // N
  float*    den  = (float*)(mEnc + N);       // N
  float*    acc  = den + N;                  // N*16
  float*    in2  = acc + (size_t)N * 16;     // N*32  -> total N*68 floats (~27 MB)

  const int EN = E + N;
  dim3 bN(256), gN((N + 255) / 256);
  dim3 bE(256), gE((EN + 255) / 256);
  const int tiles = (N + 15) / 16;
  float* outp = (float*)d_out;

  // ---- layer 1: x (N x 16) ----
  init_nodes<<<gN, bN, 0, stream>>>(mEnc, den, acc, 16, N);
  gemm16_wmma<16><<<tiles, 32, 0, stream>>>(x, W1, H, N);
  node_scores16<<<gN, bN, 0, stream>>>(H, as1, ad1, es, ed, N);
  edge_max<<<gE, bE, 0, stream>>>(ei, E, N, es, ed, mEnc);
  edge_sum<<<gE, bE, 0, stream>>>(ei, E, N, es, ed, mEnc, den);
  edge_scatter16<<<gE, bE, 0, stream>>>(ei, E, N, es, ed, mEnc, den, H, acc);
  finalize16<<<gN, bN, 0, stream>>>(acc, b1, x, in2, N);

  // ---- layers 2 & 3: [xa | x] (N x 32) ----
  const float* Ws[2]  = {W2, W3};
  const float* ass[2] = {as2, as3};
  const float* ads[2] = {ad2, ad3};
  const float* bs[2]  = {b2, b3};
  for (int l = 0; l < 2; ++l) {
    init_nodes<<<gN, bN, 0, stream>>>(mEnc, den, acc, 16, N);
    gemm16_wmma<32><<<tiles, 32, 0, stream>>>(in2, Ws[l], H, N);
    node_scores16<<<gN, bN, 0, stream>>>(H, ass[l], ads[l], es, ed, N);
    edge_max<<<gE, bE, 0, stream>>>(ei, E, N, es, ed, mEnc);
    edge_sum<<<gE, bE, 0, stream>>>(ei, E, N, es, ed, mEnc, den);
    edge_scatter16<<<gE, bE, 0, stream>>>(ei, E, N, es, ed, mEnc, den, H, acc);
    finalize16<<<gN, bN, 0, stream>>>(acc, bs[l], x, in2, N);
  }

  // ---- p head: [xa3 | x] (N x 32) -> 1 ----
  init_nodes<<<gN, bN, 0, stream>>>(mEnc, den, acc, 1, N);
  node_hp<<<gN, bN, 0, stream>>>(in2, Wp, asp, adp, H, es, ed, N);
  edge_max<<<gE, bE, 0, stream>>>(ei, E, N, es, ed, mEnc);
  edge_sum<<<gE, bE, 0, stream>>>(ei, E, N, es, ed, mEnc, den);
  edge_scatter1<<<gE, bE, 0, stream>>>(ei, E, N, es, ed, mEnc, den, H, acc);
  finalize_p<<<gN, bN, 0, stream>>>(acc, bp, outp, N);

  // ---- value head over xa3 (held in in2 cols 0..15) ----
  if (G > 0)
    value_head<<<G, 256, 0, stream>>>(in2, f1w, f1b, f2w, f2b, outp + N, P);
}